// LabelContradictionPenalty_6743098655300
// MI455X (gfx1250) — compile-verified
//
#include <hip/hip_runtime.h>
#include <hip/hip_bf16.h>
#include <cstdint>

// Problem constants from the reference (B=1024, L=20000, P=5000, C=16).
#define L_DIM   20000   // row length; 20000 f32 = 80000 B -> fits in 320KB WGP LDS
#define C_DIM   16
#define BLOCK   256     // 8 wave32 waves
#define MARGIN_F 0.05f

// ---- CDNA5 async global->LDS path (probe via __has_builtin) ----------------
#if defined(__has_builtin)
#  if __has_builtin(__builtin_amdgcn_global_load_async_to_lds_b128)
#    define HAVE_ASYNC_LDS 1
#  endif
#  if __has_builtin(__builtin_amdgcn_s_wait_asynccnt)
#    define HAVE_WAIT_ASYNC 1
#  endif
#endif
#ifndef HAVE_ASYNC_LDS
#  define HAVE_ASYNC_LDS 0
#endif
#ifndef HAVE_WAIT_ASYNC
#  define HAVE_WAIT_ASYNC 0
#endif

#if HAVE_ASYNC_LDS
// Builtin signature (from hipcc diagnostic): first param is
// 'int __attribute__((vector_size(16))) __device__ *' (AS1, non-const, v4i).
typedef int v4i __attribute__((ext_vector_type(4)));
typedef __attribute__((address_space(1))) v4i* gbl_v4i_ptr;
typedef __attribute__((address_space(3))) v4i* lds_v4i_ptr;
#endif

// Kernel 1: one workgroup per batch row.
//   Stage preds[b, 0:L] into LDS (async data-mover path), then each thread
//   gathers parent + 16 children from LDS for a strided subset of parents.
//   Deterministic per-block tree reduction -> partial[b].
__global__ __launch_bounds__(BLOCK)
void lcp_row_kernel(const float* __restrict__ preds,
                    const int*   __restrict__ parents,
                    const int*   __restrict__ children,
                    float*       __restrict__ partial,
                    int P)
{
    __shared__ float row[L_DIM];
    __shared__ float red[BLOCK];

    const int tid = threadIdx.x;
    const int b   = blockIdx.x;
    const float* __restrict__ rowg = preds + (size_t)b * (size_t)L_DIM;

    // ---- Stage the 80KB row into LDS ----
#if HAVE_ASYNC_LDS
    // 5000 x 16B chunks; each lane DMAs one b128 chunk per iteration.
    for (int chunk = tid; chunk < (L_DIM / 4); chunk += BLOCK) {
        __builtin_amdgcn_global_load_async_to_lds_b128(
            (gbl_v4i_ptr)(rowg + (size_t)chunk * 4),
            (lds_v4i_ptr)(&row[chunk * 4]),
            /*offset=*/0, /*cpol=*/0);
    }
#  if HAVE_WAIT_ASYNC
    __builtin_amdgcn_s_wait_asynccnt(0);
#  else
    asm volatile("s_wait_asynccnt 0" ::: "memory");
#  endif
#else
    // Fallback: vector loads + ds stores (compiler-generated).
    for (int chunk = tid; chunk < (L_DIM / 4); chunk += BLOCK) {
        const float4 v = reinterpret_cast<const float4*>(rowg)[chunk];
        reinterpret_cast<float4*>(row)[chunk] = v;
    }
#endif
    __syncthreads();

    // ---- Gather + hinge-penalty accumulate ----
    float sum = 0.0f;
    for (int p = tid; p < P; p += BLOCK) {
        const int parent = parents[p];
        const float ps = row[parent];

        // 16 child indices = 64B contiguous -> 4 coalesced int4 loads (L2-hot).
        const int4* __restrict__ cp =
            reinterpret_cast<const int4*>(children) + (size_t)p * 4;
        float cmax = -3.402823466e+38f;
#pragma unroll
        for (int q = 0; q < 4; ++q) {
            const int4 c4 = cp[q];
            cmax = fmaxf(cmax, row[c4.x]);
            cmax = fmaxf(cmax, row[c4.y]);
            cmax = fmaxf(cmax, row[c4.z]);
            cmax = fmaxf(cmax, row[c4.w]);
        }
        const float c = fabsf(ps - cmax);
        sum += (c > MARGIN_F) ? c : 0.0f;
    }

    // ---- Deterministic block reduction ----
    red[tid] = sum;
    __syncthreads();
#pragma unroll
    for (int s = BLOCK / 2; s > 0; s >>= 1) {
        if (tid < s) red[tid] += red[tid + s];
        __syncthreads();
    }
    if (tid == 0) partial[b] = red[0];
}

// Kernel 2: single-block deterministic reduction of B partials -> scalar.
__global__ __launch_bounds__(BLOCK)
void lcp_reduce_kernel(const float* __restrict__ partial,
                       float*       __restrict__ out,
                       int B)
{
    __shared__ float red[BLOCK];
    const int tid = threadIdx.x;
    float s = 0.0f;
    for (int i = tid; i < B; i += BLOCK) s += partial[i];
    red[tid] = s;
    __syncthreads();
#pragma unroll
    for (int k = BLOCK / 2; k > 0; k >>= 1) {
        if (tid < k) red[tid] += red[tid + k];
        __syncthreads();
    }
    if (tid == 0) out[0] = red[0] * (1.0f / (float)B);  // PENALTY_WEIGHT == 1
}

extern "C" void kernel_launch(void* const* d_in, const int* in_sizes, int n_in,
                              void* d_out, int out_size, void* d_ws, size_t ws_size,
                              hipStream_t stream)
{
    const float* preds    = (const float*)d_in[0];
    const int*   parents  = (const int*)d_in[1];   // integer inputs -> int32 per harness
    const int*   children = (const int*)d_in[2];

    const int P = in_sizes[1];
    const int B = in_sizes[0] / L_DIM;

    float* partial = (float*)d_ws;   // B floats of scratch

    lcp_row_kernel<<<B, BLOCK, 0, stream>>>(preds, parents, children, partial, P);
    lcp_reduce_kernel<<<1, BLOCK, 0, stream>>>(partial, (float*)d_out, B);
}